// LossYOLO_v3_8761733284309
// MI455X (gfx1250) — compile-verified
//
#include <hip/hip_runtime.h>
#include <hip/hip_bf16.h>

#define BATCH 256
#define NGT 8
#define HWA 10647
#define NCLS 3
#define MAXENT 24

__constant__ float c_anc[9][2] = {
    {0.0276f, 0.0361f}, {0.0386f, 0.0697f}, {0.0795f, 0.0553f},
    {0.0721f, 0.1466f}, {0.1495f, 0.1087f}, {0.1421f, 0.2861f},
    {0.2788f, 0.2163f}, {0.3750f, 0.4760f}, {0.8970f, 0.7840f}};
__constant__ int c_grids[3] = {52, 26, 13};
__constant__ int c_ceng[3] = {0, 8112, 10140};

struct BInfo {
  int count;
  int npos;
  float posLoss;   // (conf_pos + cls + txty + twth) / nums_pos, per batch
  int idx[MAXENT];
};

__device__ __forceinline__ float sigf(float x) { return 1.0f / (1.0f + expf(-x)); }
__device__ __forceinline__ float clampp(float p) {
  return fminf(fmaxf(p, 1e-6f), 1.0f - 1e-6f);
}
__device__ __forceinline__ float bce(float p, float g) {
  p = clampp(p);
  return -(g * logf(p) + (1.0f - g) * log1pf(-p));
}

// ---------------- Kernel A: match scan replay + positive-cell losses ----------------
__global__ void matchKernel(const float* __restrict__ pconf,
                            const float* __restrict__ pcls,
                            const float* __restrict__ ptxywh,
                            const float* __restrict__ gboxes,
                            const int* __restrict__ glabels,
                            BInfo* __restrict__ info) {
  int b = blockIdx.x * blockDim.x + threadIdx.x;
  if (b >= BATCH) return;

  int eidx[MAXENT];
  signed char epos[MAXENT];
  signed char egt[MAXENT];
  int n = 0;

  int   gidx[NGT];
  float gtx[NGT], gty[NGT], gtw[NGT], gth[NGT], gwt[NGT];
  int   glab[NGT];

  for (int j = 0; j < NGT; ++j) {
    const float* gb = gboxes + ((size_t)b * NGT + j) * 4;
    float l = gb[0], t = gb[1], r = gb[2], btm = gb[3];
    float cx = (l + r) * 0.5f, cy = (t + btm) * 0.5f;
    float w = r - l, h = btm - t;
    float areag = w * h;
    int best = 0; float bestiou = -1.0f;
    for (int a = 0; a < 9; ++a) {
      float aw = c_anc[a][0], ah = c_anc[a][1];
      float inter = fminf(w, aw) * fminf(h, ah);
      float iou = inter / (areag + aw * ah - inter);
      if (iou > bestiou) { bestiou = iou; best = a; }
    }
    int ceng = best / 3, aoff = best % 3;
    int grid = c_grids[ceng];
    float gf = (float)grid;
    int col = (int)(cx * gf);
    int row = (int)(cy * gf);
    int idz = c_ceng[ceng] + (row * grid + col) * 3 + aoff;
    gidx[j] = idz;
    gtx[j] = (cx - (float)col / gf) * gf;
    gty[j] = (cy - (float)row / gf) * gf;
    gtw[j] = logf(w / c_anc[best][0]);
    gth[j] = logf(h / c_anc[best][1]);
    gwt[j] = 2.0f - areag;
    glab[j] = glabels[b * NGT + j];

    // step: mark idz..idz+2 as ignore(-1), then set row at idz (last write wins)
    for (int d = 0; d < 3; ++d) {
      int c = idz + d;
      if (c >= HWA) continue;  // JAX drops OOB scatter updates
      int e = -1;
      for (int q = 0; q < n; ++q) if (eidx[q] == c) { e = q; break; }
      if (e < 0) { e = n++; eidx[e] = c; }
      epos[e] = 0; egt[e] = -1;
    }
    {
      int e = -1;
      for (int q = 0; q < n; ++q) if (eidx[q] == idz) { e = q; break; }
      epos[e] = 1; egt[e] = (signed char)j;
    }
  }

  int npos = 0;
  float ploss = 0.0f;
  for (int e = 0; e < n; ++e) {
    if (epos[e] != 1) continue;
    ++npos;
    int j = egt[e];
    size_t base = (size_t)b * HWA + eidx[e];
    // conf positive term: bce(sigmoid(pconf), 1)
    ploss += bce(sigf(pconf[base]), 1.0f);
    // class BCE
    int lab = glab[j] - 1;
    for (int c = 0; c < NCLS; ++c) {
      float pc = sigf(pcls[base * NCLS + c]);
      ploss += bce(pc, (c == lab) ? 1.0f : 0.0f);
    }
    // txty BCE * weight
    float wgt = gwt[j];
    float ptx = sigf(ptxywh[base * 4 + 0]);
    float pty = sigf(ptxywh[base * 4 + 1]);
    ploss += (bce(ptx, gtx[j]) + bce(pty, gty[j])) * wgt;
    // twth L2 * weight
    float dw = ptxywh[base * 4 + 2] - gtw[j];
    float dh = ptxywh[base * 4 + 3] - gth[j];
    ploss += (dw * dw + dh * dh) * wgt;
  }

  float numsPos = fmaxf((float)npos, 9.765625e-4f);
  BInfo bi;
  bi.count = n;
  bi.npos = npos;
  bi.posLoss = ploss / numsPos;
  for (int e = 0; e < MAXENT; ++e) bi.idx[e] = (e < n) ? eidx[e] : -1;
  info[b] = bi;
}

// ---------------- Kernel B: conf-BCE over all anchors + OHEM top-K in LDS ----------------
__global__ __launch_bounds__(256) void confKernel(const float* __restrict__ pconf,
                                                  const BInfo* __restrict__ info,
                                                  float* __restrict__ totals) {
  __shared__ float lz[HWA];
  __shared__ float rv[256];
  __shared__ int   ra[256];

  int b = blockIdx.x;
  int tid = threadIdx.x;
  const float* pc = pconf + (size_t)b * HWA;

  for (int i = tid; i < HWA; i += 256) {
    if (i + 4096 < HWA) __builtin_prefetch(pc + i + 4096, 0, 0);  // global_prefetch_b8
    float p = clampp(sigf(pc[i]));
    lz[i] = -log1pf(-p);  // x_bce with g == 0 (negative hypothesis)
  }
  __syncthreads();

  int cnt = info[b].count;
  if (tid < cnt) {
    int c = info[b].idx[tid];
    if (c >= 0 && c < HWA) lz[c] = 0.0f;  // mask pos | ign
  }
  __syncthreads();

  int K = info[b].npos * 3;  // num_neg = 3 * nums_pos (npos >= 1 always)
  float acc = 0.0f;
  for (int k = 0; k < K; ++k) {
    float bv = -1.0f; int bix = 0;
    for (int i = tid; i < HWA; i += 256) {
      float v = lz[i];
      if (v > bv) { bv = v; bix = i; }
    }
    rv[tid] = bv; ra[tid] = bix;
    __syncthreads();
    for (int s = 128; s > 0; s >>= 1) {
      if (tid < s && rv[tid + s] > rv[tid]) { rv[tid] = rv[tid + s]; ra[tid] = ra[tid + s]; }
      __syncthreads();
    }
    if (tid == 0) { acc += rv[0]; lz[ra[0]] = -1.0f; }
    __syncthreads();
  }

  if (tid == 0) {
    float numsPos = fmaxf((float)info[b].npos, 9.765625e-4f);
    totals[b] = info[b].posLoss + acc / numsPos;
  }
}

// ---------------- Kernel C: 256 -> 1 batch mean via V_WMMA_F32_16X16X4_F32 ----------------
typedef float v2f __attribute__((ext_vector_type(2)));
typedef float v8f __attribute__((ext_vector_type(8)));

__global__ void finishKernel(const float* __restrict__ totals, float* __restrict__ out) {
  int lane = threadIdx.x;  // launched with exactly 32 threads, EXEC all-1s for WMMA
  v8f acc = {0.f, 0.f, 0.f, 0.f, 0.f, 0.f, 0.f, 0.f};
  v2f ones; ones[0] = 1.0f; ones[1] = 1.0f;
  for (int ch = 0; ch < 4; ++ch) {
    v2f a;
    a[0] = totals[ch * 64 + lane * 2 + 0];
    a[1] = totals[ch * 64 + lane * 2 + 1];
    // D = A(16x4) * ones(4x16) + C : each D row holds the sum of 4 A elements
    acc = __builtin_amdgcn_wmma_f32_16x16x4_f32(false, a, false, ones, (short)0, acc,
                                                false, false);
  }
  float s = 0.0f;
#pragma unroll
  for (int i = 0; i < 8; ++i) s += acc[i];
  // lanes 0-15 hold row-sums M=0..7, lanes 16-31 hold M=8..15; combine halves
  float oth = __shfl_xor(s, 16, 32);
  if (lane == 0) out[0] = (s + oth) * (1.0f / 256.0f);
}

extern "C" void kernel_launch(void* const* d_in, const int* in_sizes, int n_in,
                              void* d_out, int out_size, void* d_ws, size_t ws_size,
                              hipStream_t stream) {
  const float* pconf  = (const float*)d_in[0];
  const float* pcls   = (const float*)d_in[1];
  const float* ptxywh = (const float*)d_in[2];
  const float* gboxes = (const float*)d_in[3];
  const int*   glabels = (const int*)d_in[4];

  BInfo* info = (BInfo*)d_ws;
  float* totals = (float*)((char*)d_ws + BATCH * sizeof(BInfo));

  matchKernel<<<8, 32, 0, stream>>>(pconf, pcls, ptxywh, gboxes, glabels, info);
  confKernel<<<BATCH, 256, 0, stream>>>(pconf, info, totals);
  finishKernel<<<1, 32, 0, stream>>>(totals, (float*)d_out);
}